// ReconNet_65292092834260
// MI455X (gfx1250) — compile-verified
//
#include <hip/hip_runtime.h>
#include <math.h>

typedef __attribute__((ext_vector_type(2))) float v2f;
typedef __attribute__((ext_vector_type(8))) float v8f;

#define VOXEL_SIZE 0.08f
#define CCH 24          // feature channels
#define HH 120
#define WW 160
#define HW (HH * WW)

// ---------------------------------------------------------------------------
// Kernel 0: zero the global stats accumulators (S1, S2, Npos) each launch.
// ---------------------------------------------------------------------------
__global__ void bp_init_stats(double* stats) {
    stats[0] = 0.0; stats[1] = 0.0; stats[2] = 0.0;
}

// ---------------------------------------------------------------------------
// Kernel 1: back-projection. Each wave32 handles 16 voxels.
//   WMMA f32 16x16x4 computes im_p = KRcam_rows x pts for 8 of the 9 views
//   (2 WMMAs); view 8 is 12 scalar FMAs on the low half-lanes.
//   Lane (h, s) ends up owning views {2h,2h+1} and {4+2h,5+2h} of voxel s,
//   so the bilinear gather work is split across the lane pair and merged
//   with shfl_xor(16).
// ---------------------------------------------------------------------------
__global__ __launch_bounds__(256)
void bp_backproject(const int*   __restrict__ coords,   // (N,4) int32
                    const float* __restrict__ origin,   // (3,)
                    const float* __restrict__ feats,    // (V,C,H,W)
                    const float* __restrict__ kr,       // (V,4,4)
                    float* __restrict__ vol,            // (N, C+1)
                    float* __restrict__ cntout,         // (N,)
                    float* __restrict__ zvox,           // (N,) per-voxel mean depth
                    double* __restrict__ stats,         // [S1, S2, Npos]
                    int N)
{
    __shared__ float sh[3][8];

    const int lane = threadIdx.x & 31;
    const int wid  = threadIdx.x >> 5;          // wave in block (8 waves)
    const int sub  = lane & 15;                 // voxel slot 0..15
    const int half = lane >> 4;                 // 0 = low half, 1 = high half
    const int voxBase = (blockIdx.x * 8 + wid) * 16;
    const int vox  = voxBase + sub;
    const int voxc = vox < N ? vox : (N - 1);   // clamp for safe loads

    // ---- world coordinates of this lane's voxel (both halves compute it) ----
    const float ox = origin[0], oy = origin[1], oz = origin[2];
    const float wx = (float)coords[voxc * 4 + 1] * VOXEL_SIZE + ox;
    const float wy = (float)coords[voxc * 4 + 2] * VOXEL_SIZE + oy;
    const float wz = (float)coords[voxc * 4 + 3] * VOXEL_SIZE + oz;

    // ---- B matrix (4x16): column n = homogeneous point of voxel n ----
    // lane n: K=0,1 -> (x, y); lane n+16: K=2,3 -> (z, 1)
    v2f b;
    b[0] = half ? wz : wx;
    b[1] = half ? 1.0f : wy;

    // ---- A matrices (16x4): row m = one (view,row) of KRcam ----
    // rows 0-5 = views {base,base+1} x/y/z ; rows 6,7 zero ; rows 8-13 = views
    // {base+2,base+3} ; rows 14,15 zero.  Lane m holds K=0,1; lane m+16 K=2,3.
    v2f a1; a1[0] = 0.f; a1[1] = 0.f;
    v2f a2; a2[0] = 0.f; a2[1] = 0.f;
    {
        const int blk = sub >> 3;       // which 8-row block
        const int lr  = sub & 7;        // row within block
        if (lr < 6) {
            const int vloc = lr / 3;    // 0 or 1 (view within block)
            const int row  = lr - vloc * 3;
            const int view = blk * 2 + vloc;
            const float* p1 = kr + (view    ) * 16 + row * 4 + half * 2;
            const float* p2 = kr + (view + 4) * 16 + row * 4 + half * 2;
            a1[0] = p1[0]; a1[1] = p1[1];
            a2[0] = p2[0]; a2[1] = p2[1];
        }
    }

    // ---- projections: D[comp][voxel] = KRcam row . pts ----
    v8f zero8 = {};
    v8f d1 = __builtin_amdgcn_wmma_f32_16x16x4_f32(
        false, a1, false, b, (short)0, zero8, false, false);
    v8f d2 = __builtin_amdgcn_wmma_f32_16x16x4_f32(
        false, a2, false, b, (short)0, zero8, false, false);

    // ---- per-lane view accumulation (bilinear gather) ----
    float acc[CCH];
    #pragma unroll
    for (int c = 0; c < CCH; ++c) acc[c] = 0.f;
    float zsum = 0.f, cnt = 0.f;

    auto sampleView = [&](int v, float px, float py, float pz) {
        const float zs   = fabsf(pz) > 1e-6f ? pz : 1e-6f;
        const float invz = 1.0f / zs;          // one precise divide per view
        const float x = px * invz;
        const float y = py * invz;
        const bool m = (x >= 0.f) && (x <= (float)(WW - 1)) &&
                       (y >= 0.f) && (y <= (float)(HH - 1)) && (pz > 0.f);
        if (m) {
            const float fx0 = floorf(x), fy0 = floorf(y);
            const int ix0 = (int)fx0, iy0 = (int)fy0;
            const float tx = x - fx0, ty = y - fy0;
            const int ix1 = min(ix0 + 1, WW - 1);
            const int iy1 = min(iy0 + 1, HH - 1);
            const float w00 = (1.f - tx) * (1.f - ty);
            const float w10 = tx * (1.f - ty);
            const float w01 = (1.f - tx) * ty;
            const float w11 = tx * ty;
            const int o00 = iy0 * WW + ix0, o10 = iy0 * WW + ix1;
            const int o01 = iy1 * WW + ix0, o11 = iy1 * WW + ix1;
            const float* base = feats + (size_t)v * CCH * HW;
            #pragma unroll
            for (int c = 0; c < CCH; ++c) {
                const float* p = base + c * HW;
                acc[c] += w00 * p[o00] + w10 * p[o10]
                        + w01 * p[o01] + w11 * p[o11];
            }
            zsum += pz;
            cnt  += 1.f;
        }
    };

    // D layout: lane half h sees rows (8h..8h+7) in VGPRs 0..7
    sampleView(2 * half + 0, d1[0], d1[1], d1[2]);
    sampleView(2 * half + 1, d1[3], d1[4], d1[5]);
    sampleView(4 + 2 * half + 0, d2[0], d2[1], d2[2]);
    sampleView(4 + 2 * half + 1, d2[3], d2[4], d2[5]);
    if (half == 0) {                       // view 8: plain VALU projection
        const float* m8 = kr + 8 * 16;
        const float px = m8[0] * wx + m8[1] * wy + m8[2]  * wz + m8[3];
        const float py = m8[4] * wx + m8[5] * wy + m8[6]  * wz + m8[7];
        const float pz = m8[8] * wx + m8[9] * wy + m8[10] * wz + m8[11];
        sampleView(8, px, py, pz);
    }

    // ---- merge the two halves of each lane pair ----
    #pragma unroll
    for (int c = 0; c < CCH; ++c) acc[c] += __shfl_xor(acc[c], 16, 32);
    zsum += __shfl_xor(zsum, 16, 32);
    cnt  += __shfl_xor(cnt,  16, 32);

    // ---- low half writes results ----
    float zm = 0.f;
    if (half == 0 && vox < N) {
        const float denom = fmaxf(cnt, 1.0f);
        const float inv = 1.0f / denom;
        float* row = vol + (size_t)vox * (CCH + 1);
        #pragma unroll
        for (int c = 0; c < CCH; ++c) row[c] = acc[c] * inv;
        cntout[vox] = cnt;
        zm = zsum * inv;
        zvox[vox] = zm;
    }

    // ---- block-level reduction of z statistics ----
    float s1 = (half == 0 && vox < N && zm > 0.f) ? zm  : 0.f;
    float s2 = s1 * zm;
    float np = (half == 0 && vox < N && zm > 0.f) ? 1.f : 0.f;
    #pragma unroll
    for (int off = 16; off >= 1; off >>= 1) {
        s1 += __shfl_xor(s1, off, 32);
        s2 += __shfl_xor(s2, off, 32);
        np += __shfl_xor(np, off, 32);
    }
    if (lane == 0) { sh[0][wid] = s1; sh[1][wid] = s2; sh[2][wid] = np; }
    __syncthreads();
    if (threadIdx.x == 0) {
        float t1 = 0.f, t2 = 0.f, t3 = 0.f;
        #pragma unroll
        for (int w = 0; w < 8; ++w) { t1 += sh[0][w]; t2 += sh[1][w]; t3 += sh[2][w]; }
        atomicAdd(&stats[0], (double)t1);
        atomicAdd(&stats[1], (double)t2);
        atomicAdd(&stats[2], (double)t3);
    }
}

// ---------------------------------------------------------------------------
// Kernel 2: finalize z mean / inverse std from the global accumulators.
//   sum((z-m)^2) = S2 - S1^2 / n  (one-pass)
// ---------------------------------------------------------------------------
__global__ void bp_finalize_stats(const double* __restrict__ stats,
                                  float* __restrict__ ms) {
    const double S1 = stats[0], S2 = stats[1], NP = stats[2];
    const double n  = NP > 0.5 ? NP : 1.0;
    const double m  = S1 / n;
    double var = S2 - (S1 * S1) / n;
    if (var < 0.0) var = 0.0;
    const double sd = sqrt(var) + 1e-5;
    ms[0] = (float)m;
    ms[1] = (float)(1.0 / sd);
}

// ---------------------------------------------------------------------------
// Kernel 3: write the z-score-normalized depth channel (channel C).
// ---------------------------------------------------------------------------
__global__ void bp_znorm(const float* __restrict__ zvox,
                         const float* __restrict__ ms,
                         float* __restrict__ vol, int N) {
    const int i = blockIdx.x * blockDim.x + threadIdx.x;
    if (i >= N) return;
    const float zm = zvox[i];
    const float zn = zm > 0.f ? (zm - ms[0]) * ms[1] : 0.f;
    vol[(size_t)i * (CCH + 1) + CCH] = zn;
}

// ---------------------------------------------------------------------------
extern "C" void kernel_launch(void* const* d_in, const int* in_sizes, int n_in,
                              void* d_out, int out_size, void* d_ws, size_t ws_size,
                              hipStream_t stream) {
    const int*   coords = (const int*)d_in[0];
    const float* origin = (const float*)d_in[1];
    const float* feats  = (const float*)d_in[2];
    const float* kr     = (const float*)d_in[3];
    const int N = in_sizes[0] / 4;

    float* vol = (float*)d_out;                       // (N, C+1)
    float* cnt = (float*)d_out + (size_t)N * (CCH + 1);

    char*   ws    = (char*)d_ws;
    float*  zvox  = (float*)ws;                                   // N floats
    double* stats = (double*)(ws + (size_t)N * sizeof(float));    // 3 doubles
    float*  ms    = (float*)(ws + (size_t)N * sizeof(float) + 32);// 2 floats

    bp_init_stats<<<1, 1, 0, stream>>>(stats);

    const int blocks = (N + 127) / 128;               // 8 waves * 16 voxels
    bp_backproject<<<blocks, 256, 0, stream>>>(coords, origin, feats, kr,
                                               vol, cnt, zvox, stats, N);

    bp_finalize_stats<<<1, 1, 0, stream>>>(stats, ms);
    bp_znorm<<<(N + 255) / 256, 256, 0, stream>>>(zvox, ms, vol, N);
}